// SALTEdoraLinearV4_75874892251349
// MI455X (gfx1250) — compile-verified
//
#include <hip/hip_runtime.h>
#include <hip/hip_bf16.h>
#include <cstdint>

typedef __attribute__((ext_vector_type(2))) float v2f_t;
typedef __attribute__((ext_vector_type(8))) float v8f_t;

#define IN_DIM  4096
#define OUT_DIM 4096
#define RTOP    64
#define RTAIL   4032
#define RINT    4
#define MTOT    4096  // B * S

// ---------------------------------------------------------------------------
// CDNA5 async global->LDS copy (GLOBAL_LOAD_ASYNC_TO_LDS_B128, ASYNCcnt).
// Builtin signature (from compiler diagnostic): pointers to 128-bit int
// vectors in AS1 (global) / AS3 (LDS).  Falls back to a synchronous VGPR
// round-trip if the builtin is unavailable.
// ---------------------------------------------------------------------------
#if __has_builtin(__builtin_amdgcn_global_load_async_to_lds_b128)
#define ASYNC_LDS_COPY 1
#endif

typedef int v4i_vs __attribute__((vector_size(16)));
typedef __attribute__((address_space(1))) v4i_vs* as1_v4i_ptr;
typedef __attribute__((address_space(3))) v4i_vs* as3_v4i_ptr;

__device__ __forceinline__ void async_copy16(const float* g, float* l) {
#ifdef ASYNC_LDS_COPY
  __builtin_amdgcn_global_load_async_to_lds_b128(
      (as1_v4i_ptr)(uintptr_t)g,
      (as3_v4i_ptr)(uint32_t)(uintptr_t)l,
      0, 0);
#else
  *(float4*)l = *(const float4*)g;
#endif
}

__device__ __forceinline__ void async_wait0() {
#ifdef ASYNC_LDS_COPY
#if __has_builtin(__builtin_amdgcn_s_wait_asynccnt)
  __builtin_amdgcn_s_wait_asynccnt(0);
#else
  asm volatile("s_wait_asynccnt 0x0" ::: "memory");
#endif
#endif
}

// ---------------------------------------------------------------------------
// Kernel 0: tiny prep.  s_cat[r] = relu(S_top*alpha+beta) for r<64, else S_tail.
// A4 = (U_tail_r * S_tail_r * relu(D)) @ R   -> (4096, 4)
// ---------------------------------------------------------------------------
__global__ void prep_kernel(const float* __restrict__ S_top,
                            const float* __restrict__ alpha,
                            const float* __restrict__ beta,
                            const float* __restrict__ S_tail,
                            const float* __restrict__ U_tail_r,
                            const float* __restrict__ S_tail_r,
                            const float* __restrict__ D,
                            const float* __restrict__ R,
                            float* __restrict__ s_cat,
                            float* __restrict__ A4) {
  const int t = blockIdx.x * blockDim.x + threadIdx.x;
  const int stride = gridDim.x * blockDim.x;

  for (int r = t; r < IN_DIM; r += stride) {
    float s;
    if (r < RTOP) {
      float v = S_top[r] * alpha[r] + beta[r];
      s = v > 0.f ? v : 0.f;
    } else {
      s = S_tail[r - RTOP];
    }
    s_cat[r] = s;
  }

  float uscale[RINT];
#pragma unroll
  for (int j = 0; j < RINT; ++j) {
    float d = D[j];
    d = d > 0.f ? d : 0.f;
    uscale[j] = S_tail_r[j] * d;
  }

  for (int o = t; o < OUT_DIM; o += stride) {
    float a0 = 0.f, a1 = 0.f, a2 = 0.f, a3 = 0.f;
#pragma unroll
    for (int tt = 0; tt < RINT; ++tt) {
      float u = U_tail_r[o * RINT + tt] * uscale[tt];
      a0 += u * R[tt * RINT + 0];
      a1 += u * R[tt * RINT + 1];
      a2 += u * R[tt * RINT + 2];
      a3 += u * R[tt * RINT + 3];
    }
    A4[o * 4 + 0] = a0;
    A4[o * 4 + 1] = a1;
    A4[o * 4 + 2] = a2;
    A4[o * 4 + 3] = a3;
  }
}

// ---------------------------------------------------------------------------
// Kernel 1: W_eff[o][i] = sum_r U_cat[o][r] * s_cat[r] * Vh_cat[r][i]
//                        + sum_j A4[o][j] * Vh_r[j][i]      (rank-4 epilogue)
// Block tile 64x64, 128 threads (4 waves), wave tile 32x32 = 2x2 WMMA f32 tiles.
// A tile staged with async global->LDS copies; B tile staged synchronously
// because the per-row s_cat scale is applied in flight.
// ---------------------------------------------------------------------------
__launch_bounds__(128)
__global__ void weff_gemm_kernel(const float* __restrict__ U_top,
                                 const float* __restrict__ U_tail,
                                 const float* __restrict__ Vh_top,
                                 const float* __restrict__ Vh_tail,
                                 const float* __restrict__ s_cat,
                                 const float* __restrict__ A4,
                                 const float* __restrict__ Vh_r,
                                 float* __restrict__ W) {
  __shared__ __align__(16) float Asl[64][32];  // [o][r]
  __shared__ __align__(16) float Bsl[32][64];  // [r][i]

  const int tid = threadIdx.x;
  const int o0 = blockIdx.y * 64;
  const int i0 = blockIdx.x * 64;
  const int wave = tid >> 5;
  const int lane = tid & 31;
  const int wm = (wave >> 1) * 32;
  const int wn = (wave & 1) * 32;
  const int lr = lane & 15;
  const int lg = lane >> 4;  // 0 or 1

  v8f_t acc00 = {}, acc01 = {}, acc10 = {}, acc11 = {};

  for (int c = 0; c < IN_DIM / 32; ++c) {
    const int r0 = c * 32;
    const float* As;
    const float* Bs;
    int lda, ka;
    if (r0 < RTOP) {
      As = U_top;  lda = RTOP;  ka = r0;
      Bs = Vh_top + (size_t)r0 * IN_DIM;
    } else {
      As = U_tail; lda = RTAIL; ka = r0 - RTOP;
      Bs = Vh_tail + (size_t)(r0 - RTOP) * IN_DIM;
    }

    __syncthreads();
    // A tile: pure copy -> async global->LDS (no VGPR round trip).
#pragma unroll
    for (int it = 0; it < 4; ++it) {
      const int f = tid + 128 * it;
      const int row = f >> 3;
      const int cv = (f & 7) * 4;
      async_copy16(As + (size_t)(o0 + row) * lda + ka + cv, &Asl[row][cv]);
    }
    // B tile: load, scale by s_cat[r], store to LDS.
#pragma unroll
    for (int it = 0; it < 4; ++it) {
      const int f = tid + 128 * it;
      const int row = f >> 4;
      const int cv = (f & 15) * 4;
      const float s = s_cat[r0 + row];
      float4 v = *(const float4*)(Bs + (size_t)row * IN_DIM + i0 + cv);
      v.x *= s; v.y *= s; v.z *= s; v.w *= s;
      *(float4*)&Bsl[row][cv] = v;
    }
    if (c + 1 < IN_DIM / 32) {  // hint next chunk into cache
      __builtin_prefetch(Bs + (size_t)32 * IN_DIM + i0 + (tid & 15) * 4, 0, 1);
    }
    async_wait0();
    __syncthreads();

#pragma unroll
    for (int kk = 0; kk < 32; kk += 4) {
      const int kb = kk + 2 * lg;
      v2f_t a0 = *(const v2f_t*)&Asl[wm + lr][kb];
      v2f_t a1 = *(const v2f_t*)&Asl[wm + 16 + lr][kb];
      v2f_t b0, b1;
      b0.x = Bsl[kb][wn + lr];      b0.y = Bsl[kb + 1][wn + lr];
      b1.x = Bsl[kb][wn + 16 + lr]; b1.y = Bsl[kb + 1][wn + 16 + lr];
      acc00 = __builtin_amdgcn_wmma_f32_16x16x4_f32(false, a0, false, b0, (short)0, acc00, false, false);
      acc01 = __builtin_amdgcn_wmma_f32_16x16x4_f32(false, a0, false, b1, (short)0, acc01, false, false);
      acc10 = __builtin_amdgcn_wmma_f32_16x16x4_f32(false, a1, false, b0, (short)0, acc10, false, false);
      acc11 = __builtin_amdgcn_wmma_f32_16x16x4_f32(false, a1, false, b1, (short)0, acc11, false, false);
    }
  }

  // Epilogue: add rank-4 delta and store.
#pragma unroll
  for (int tm = 0; tm < 2; ++tm) {
#pragma unroll
    for (int tn = 0; tn < 2; ++tn) {
      const v8f_t& acc = tm == 0 ? (tn == 0 ? acc00 : acc01)
                                 : (tn == 0 ? acc10 : acc11);
      const int nloc = wn + 16 * tn + lr;
      const size_t i = (size_t)(i0 + nloc);
      const float vr0 = Vh_r[0 * IN_DIM + i];
      const float vr1 = Vh_r[1 * IN_DIM + i];
      const float vr2 = Vh_r[2 * IN_DIM + i];
      const float vr3 = Vh_r[3 * IN_DIM + i];
#pragma unroll
      for (int v = 0; v < 8; ++v) {
        const int mloc = wm + 16 * tm + v + 8 * lg;
        const size_t o = (size_t)(o0 + mloc);
        const float4 a4 = *(const float4*)(A4 + o * 4);
        W[o * IN_DIM + i] =
            acc[v] + a4.x * vr0 + a4.y * vr1 + a4.z * vr2 + a4.w * vr3;
      }
    }
  }
}

// ---------------------------------------------------------------------------
// Kernel 2: out[m][o] = sum_i x[m][i] * W[o][i] + bias[o]
// Both tiles are pure copies -> async global->LDS staging for A and B.
// B tile stored [n][k] so both fragment loads are contiguous ds_load_b64.
// ---------------------------------------------------------------------------
__launch_bounds__(128)
__global__ void out_gemm_kernel(const float* __restrict__ X,
                                const float* __restrict__ W,
                                const float* __restrict__ bias,
                                float* __restrict__ Out) {
  __shared__ __align__(16) float Asl[64][32];  // [m][k]
  __shared__ __align__(16) float Btl[64][32];  // [n][k]  (W row-major is already [o][i])

  const int tid = threadIdx.x;
  const int m0 = blockIdx.y * 64;
  const int n0 = blockIdx.x * 64;
  const int wave = tid >> 5;
  const int lane = tid & 31;
  const int wm = (wave >> 1) * 32;
  const int wn = (wave & 1) * 32;
  const int lr = lane & 15;
  const int lg = lane >> 4;

  v8f_t acc00 = {}, acc01 = {}, acc10 = {}, acc11 = {};

  for (int c = 0; c < IN_DIM / 32; ++c) {
    const int k0 = c * 32;
    __syncthreads();
#pragma unroll
    for (int it = 0; it < 4; ++it) {
      const int f = tid + 128 * it;
      const int row = f >> 3;
      const int cv = (f & 7) * 4;
      async_copy16(X + (size_t)(m0 + row) * IN_DIM + k0 + cv, &Asl[row][cv]);
      async_copy16(W + (size_t)(n0 + row) * IN_DIM + k0 + cv, &Btl[row][cv]);
    }
    if (c + 1 < IN_DIM / 32) {
      __builtin_prefetch(X + (size_t)(m0 + (tid >> 1)) * IN_DIM + k0 + 32, 0, 1);
    }
    async_wait0();
    __syncthreads();

#pragma unroll
    for (int kk = 0; kk < 32; kk += 4) {
      const int kb = kk + 2 * lg;
      v2f_t a0 = *(const v2f_t*)&Asl[wm + lr][kb];
      v2f_t a1 = *(const v2f_t*)&Asl[wm + 16 + lr][kb];
      v2f_t b0 = *(const v2f_t*)&Btl[wn + lr][kb];
      v2f_t b1 = *(const v2f_t*)&Btl[wn + 16 + lr][kb];
      acc00 = __builtin_amdgcn_wmma_f32_16x16x4_f32(false, a0, false, b0, (short)0, acc00, false, false);
      acc01 = __builtin_amdgcn_wmma_f32_16x16x4_f32(false, a0, false, b1, (short)0, acc01, false, false);
      acc10 = __builtin_amdgcn_wmma_f32_16x16x4_f32(false, a1, false, b0, (short)0, acc10, false, false);
      acc11 = __builtin_amdgcn_wmma_f32_16x16x4_f32(false, a1, false, b1, (short)0, acc11, false, false);
    }
  }

#pragma unroll
  for (int tm = 0; tm < 2; ++tm) {
#pragma unroll
    for (int tn = 0; tn < 2; ++tn) {
      const v8f_t& acc = tm == 0 ? (tn == 0 ? acc00 : acc01)
                                 : (tn == 0 ? acc10 : acc11);
      const int nloc = wn + 16 * tn + lr;
      const size_t n = (size_t)(n0 + nloc);
      const float b = bias[n];
#pragma unroll
      for (int v = 0; v < 8; ++v) {
        const int mloc = wm + 16 * tm + v + 8 * lg;
        Out[(size_t)(m0 + mloc) * OUT_DIM + n] = acc[v] + b;
      }
    }
  }
}

// ---------------------------------------------------------------------------
extern "C" void kernel_launch(void* const* d_in, const int* in_sizes, int n_in,
                              void* d_out, int out_size, void* d_ws, size_t ws_size,
                              hipStream_t stream) {
  const float* x         = (const float*)d_in[0];
  const float* U_top     = (const float*)d_in[1];
  const float* S_top     = (const float*)d_in[2];
  const float* Vh_top    = (const float*)d_in[3];
  const float* U_tail    = (const float*)d_in[4];
  const float* S_tail    = (const float*)d_in[5];
  const float* Vh_tail   = (const float*)d_in[6];
  const float* U_tail_r  = (const float*)d_in[7];
  const float* S_tail_r  = (const float*)d_in[8];
  const float* Vh_tail_r = (const float*)d_in[9];
  const float* alpha     = (const float*)d_in[10];
  const float* beta      = (const float*)d_in[11];
  const float* D         = (const float*)d_in[12];
  const float* R         = (const float*)d_in[13];
  const float* bias      = (const float*)d_in[14];

  float* ws    = (float*)d_ws;
  float* s_cat = ws;             // 4096 floats
  float* A4    = ws + 4096;      // 4096*4 floats
  float* W     = ws + 20480;     // 4096*4096 floats (~64 MB)

  prep_kernel<<<16, 256, 0, stream>>>(S_top, alpha, beta, S_tail,
                                      U_tail_r, S_tail_r, D, R, s_cat, A4);

  dim3 g1(IN_DIM / 64, OUT_DIM / 64);
  weff_gemm_kernel<<<g1, 128, 0, stream>>>(U_top, U_tail, Vh_top, Vh_tail,
                                           s_cat, A4, Vh_tail_r, W);

  dim3 g2(OUT_DIM / 64, MTOT / 64);
  out_gemm_kernel<<<g2, 128, 0, stream>>>(x, W, bias, (float*)d_out);
}